// YoloLayer_86388972192169
// MI455X (gfx1250) — compile-verified
//
#include <hip/hip_runtime.h>
#include <cstdint>
#include <cstddef>

// ---------------------------------------------------------------------------
// YOLO layer: channel-major -> channel-minor transpose fused with
// sigmoid/exp/affine transforms.  Pure streaming op: ~353 MB of traffic,
// ~15 us floor at 23.3 TB/s.  One workgroup per (b, a, gy) slab.
// Phase 1 staging uses the CDNA5 Tensor Data Mover (one 2D-tile descriptor
// per workgroup) when available, else async global->LDS b32, else b128 NT.
// ---------------------------------------------------------------------------

#define NBATCH   64
#define NANCH    3
#define NCLS     80
#define NGRID    52
#define CHN      (5 + NCLS)        // 85
#define PLANE    (NGRID * NGRID)   // 2704 floats per channel plane
#define ROWF     (NGRID * CHN)     // 4420 floats per (b,a,gy) slab
#define VEC4     (ROWF / 4)        // 1105 float4s
#define QPC      (NGRID / 4)       // 13 float4s per channel segment
#define STRIDE_F 8.0f              // 416 / 52

typedef float        v4f   __attribute__((ext_vector_type(4)));
typedef unsigned int u32;
typedef u32          u32x4 __attribute__((ext_vector_type(4)));
typedef int          i32x4 __attribute__((ext_vector_type(4)));
typedef int          i32x8 __attribute__((ext_vector_type(8)));

#if defined(__gfx1250__) && \
    __has_builtin(__builtin_amdgcn_tensor_load_to_lds) && \
    __has_builtin(__builtin_amdgcn_s_wait_tensorcnt)
#define USE_TDM 1
#elif defined(__gfx1250__) && \
    __has_builtin(__builtin_amdgcn_global_load_async_to_lds_b32) && \
    __has_builtin(__builtin_amdgcn_s_wait_asynccnt)
#define USE_ASYNC_LDS 1
typedef __attribute__((address_space(1))) int* gptr_i32;
typedef __attribute__((address_space(3))) int* lptr_i32;
#endif

// TDM stores rows packed (pitch = tile_dim0 = 52); padded pitch 53 otherwise.
#if defined(USE_TDM)
#define LDS_PITCH 52
#else
#define LDS_PITCH 53
#endif

__device__ __forceinline__ float fsigmoid(float v) {
    // 1 / (1 + e^-v) via hardware v_exp_f32 + fast reciprocal
    return __fdividef(1.0f, 1.0f + __expf(-v));
}

__global__ __launch_bounds__(256) void yolo_layer_kernel(
    const float* __restrict__ x,        // (64, 255, 52, 52)
    const float* __restrict__ anchors,  // (3, 2)
    float* __restrict__ out)            // (64, 3*52*52, 85)
{
    __shared__ float tile[CHN * LDS_PITCH];

    const int tid = threadIdx.x;
    int blk = (int)blockIdx.x;
    const int gy = blk % NGRID;  blk /= NGRID;
    const int a  = blk % NANCH;
    const int b  = blk / NANCH;

    const float aw = anchors[2 * a + 0];
    const float ah = anchors[2 * a + 1];

    // Input slab base: element (c, gx) lives at src + c*PLANE + gx
    const float* __restrict__ src =
        x + (size_t)(b * (NANCH * CHN) + a * CHN) * PLANE + (size_t)gy * NGRID;

    // ---- Phase 1: stage slab into LDS as [c][gx] --------------------------
#if defined(USE_TDM)
    if (tid < 32) {  // one wave issues the DMA descriptor
        const unsigned long long gaddr = (unsigned long long)(size_t)src;
        const u32 laddr = (u32)(size_t)((__attribute__((address_space(3))) char*)tile);

        const u32 TD0 = (u32)PLANE;   // tensor dim0 length  (no OOB clipping)
        const u32 TD1 = (u32)CHN;     // tensor dim1 length

        // D# group 0: count=1 | lds_addr | global_addr[56:0] | type=2
        u32x4 g0;
        g0[0] = 1u;                                    // count=1, user mode
        g0[1] = laddr;                                 // LDS byte address
        g0[2] = (u32)gaddr;                            // global addr [31:0]
        g0[3] = (u32)((gaddr >> 32) & 0x01FFFFFFu)     // global addr [56:32]
              | (2u << 30);                            // type = 2 ("image")

        // D# group 1: data_size=4B, dims/strides, tile 52 x 85
        i32x8 g1;
        g1[0] = (int)(2u << 16);                       // data_size=2 -> 4 bytes
        g1[1] = (int)((TD0 & 0xFFFFu) << 16);          // tensor_dim0[15:0]
        g1[2] = (int)((TD0 >> 16) | ((TD1 & 0xFFFFu) << 16));
        g1[3] = (int)((TD1 >> 16) | ((u32)NGRID << 16));   // tile_dim0 = 52
        g1[4] = (int)CHN;                              // tile_dim1=85, tile_dim2=0
        g1[5] = (int)PLANE;                            // tensor_dim0_stride = 2704
        g1[6] = 0;
        g1[7] = 0;

        i32x4 g2 = {0, 0, 0, 0};                       // dims 2+ unused (2D tile)
        i32x4 g3 = {0, 0, 0, 0};

#if defined(__clang_major__) && (__clang_major__ >= 23)
        i32x8 g4 = {0, 0, 0, 0, 0, 0, 0, 0};
        __builtin_amdgcn_tensor_load_to_lds(g0, g1, g2, g3, g4, 0);
#else
        __builtin_amdgcn_tensor_load_to_lds(g0, g1, g2, g3, 0);
#endif
        __builtin_amdgcn_s_wait_tensorcnt(0);
    }
    __syncthreads();
#elif defined(USE_ASYNC_LDS)
    for (int i = tid; i < ROWF; i += 256) {
        const int c  = i / NGRID;
        const int gx = i - c * NGRID;
        __builtin_amdgcn_global_load_async_to_lds_b32(
            (gptr_i32)(src + c * PLANE + gx),
            (lptr_i32)&tile[c * LDS_PITCH + gx],
            0, 0);
    }
    __builtin_amdgcn_s_wait_asynccnt(0);
    __syncthreads();
#else
    for (int i = tid; i < VEC4; i += 256) {
        const int c = i / QPC;
        const int q = (i - c * QPC) * 4;
        v4f v = __builtin_nontemporal_load((const v4f*)(src + c * PLANE + q));
        const int l = c * LDS_PITCH + q;
        tile[l + 0] = v.x;
        tile[l + 1] = v.y;
        tile[l + 2] = v.z;
        tile[l + 3] = v.w;
    }
    __syncthreads();
#endif

    // ---- Phase 2: transform + contiguous 16B NT stores --------------------
    // Output slab is a fully contiguous 4420-float region (16B aligned).
    float* __restrict__ dst =
        out + (size_t)((b * NANCH + a) * PLANE + gy * NGRID) * CHN;

    for (int j = tid; j < VEC4; j += 256) {
        const int o = 4 * j;
        int gx = o / CHN;           // single division per float4
        int c  = o - gx * CHN;      // then carry (c,gx) incrementally
        v4f r;
#pragma unroll
        for (int t = 0; t < 4; ++t) {
            const float v = tile[c * LDS_PITCH + gx];
            float rv;
            if (c >= 4) {                     // conf + 80 classes: sigmoid
                rv = fsigmoid(v);
            } else if (c == 0) {              // (sigmoid(x)+gx)*stride
                rv = (fsigmoid(v) + (float)gx) * STRIDE_F;
            } else if (c == 1) {              // (sigmoid(y)+gy)*stride
                rv = (fsigmoid(v) + (float)gy) * STRIDE_F;
            } else if (c == 2) {              // exp(w)*anchor_w
                rv = __expf(v) * aw;
            } else {                          // exp(h)*anchor_h
                rv = __expf(v) * ah;
            }
            r[t] = rv;
            if (++c == CHN) { c = 0; ++gx; }
        }
        __builtin_nontemporal_store(r, (v4f*)(dst + o));
    }
}

extern "C" void kernel_launch(void* const* d_in, const int* in_sizes, int n_in,
                              void* d_out, int out_size, void* d_ws, size_t ws_size,
                              hipStream_t stream) {
    (void)in_sizes; (void)n_in; (void)out_size; (void)d_ws; (void)ws_size;
    const float* x       = (const float*)d_in[0];
    const float* anchors = (const float*)d_in[1];
    float*       out     = (float*)d_out;

    const int nblocks = NBATCH * NANCH * NGRID;   // 64*3*52 = 9984
    yolo_layer_kernel<<<dim3(nblocks), dim3(256), 0, stream>>>(x, anchors, out);
}